// SimpleUpscaler_86964497809437
// MI455X (gfx1250) — compile-verified
//
#include <hip/hip_runtime.h>

typedef float v4f __attribute__((ext_vector_type(4)));

#define NUM_CLASSES 1000
#define EMBED 512
#define SEQ 4096
#define BATCH 16

// emb[s,e] = W[e, idx[s]] + b[e]; out[bi,s,e] = src[bi,s,e] + emb[s,e]
// One thread per float4 of the [SEQ, EMBED] embedding plane; batch loop in regs.
__global__ __launch_bounds__(256) void knowledge_embed_add_kernel(
    const int*   __restrict__ idx,   // [SEQ]
    const float* __restrict__ src,   // [BATCH, SEQ, EMBED]
    const float* __restrict__ W,     // [EMBED, NUM_CLASSES] row-major
    const float* __restrict__ b,     // [EMBED]
    float*       __restrict__ out)   // [BATCH, SEQ, EMBED]
{
    const int E4 = EMBED / 4;                       // 128 float4 per row
    const int i  = blockIdx.x * 256 + threadIdx.x;  // vec4 index over SEQ*E4
    if (i >= SEQ * E4) return;

    const int s = i >> 7;            // i / 128  (wave-uniform)
    const int e = (i & 127) << 2;    // element offset within embedding row
    const int c = idx[s];            // gathered class id (uniform per wave)

    // b: small + hot -> regular temporal load (stays in cache)
    const v4f bv = *reinterpret_cast<const v4f*>(b + e);

    // Strided gather down a column of W (stride = NUM_CLASSES floats).
    // W is only 2 MB -> resident in L2; keep temporal policy.
    v4f emb;
    emb.x = W[(size_t)(e + 0) * NUM_CLASSES + c] + bv.x;
    emb.y = W[(size_t)(e + 1) * NUM_CLASSES + c] + bv.y;
    emb.z = W[(size_t)(e + 2) * NUM_CLASSES + c] + bv.z;
    emb.w = W[(size_t)(e + 3) * NUM_CLASSES + c] + bv.w;

    const size_t off    = (size_t)i * 4;
    const size_t stride = (size_t)SEQ * EMBED;      // 2M floats per batch slice

    // 268 MB of pure streaming: non-temporal so we don't evict W/b from L2.
    // Full unroll -> 16 independent NT b128 loads in flight per thread.
#pragma unroll
    for (int bi = 0; bi < BATCH; ++bi) {
        const v4f sv = __builtin_nontemporal_load(
            reinterpret_cast<const v4f*>(src + (size_t)bi * stride + off));
        v4f ov;
        ov.x = sv.x + emb.x;
        ov.y = sv.y + emb.y;
        ov.z = sv.z + emb.z;
        ov.w = sv.w + emb.w;
        __builtin_nontemporal_store(ov,
            reinterpret_cast<v4f*>(out + (size_t)bi * stride + off));
    }
}

extern "C" void kernel_launch(void* const* d_in, const int* in_sizes, int n_in,
                              void* d_out, int out_size, void* d_ws, size_t ws_size,
                              hipStream_t stream) {
    (void)in_sizes; (void)n_in; (void)d_ws; (void)ws_size; (void)out_size;

    const int*   idx = (const int*)  d_in[0];  // knowledge_input [SEQ]
    const float* src = (const float*)d_in[1];  // [BATCH, SEQ, EMBED]
    const float* W   = (const float*)d_in[2];  // [EMBED, NUM_CLASSES]
    const float* b   = (const float*)d_in[3];  // [EMBED]
    float*       out = (float*)d_out;          // [BATCH, SEQ, EMBED]

    const int total_vec4 = SEQ * (EMBED / 4);  // 524288 threads
    const int block = 256;
    const int grid  = (total_vec4 + block - 1) / block;  // 2048

    knowledge_embed_add_kernel<<<grid, block, 0, stream>>>(idx, src, W, b, out);
}